// PointAttentionLayer_62062277427289
// MI455X (gfx1250) — compile-verified
//
#include <hip/hip_runtime.h>
#include <math.h>

typedef __attribute__((ext_vector_type(2))) float v2f;
typedef __attribute__((ext_vector_type(8))) float v8f;

#define HEADS 8
#define HD 8
#define KS 7
#define CH 64
#define C3 192
#define HW 48
#define IMGPIX 2304          // 48*48
#define NPIX 36864           // 2*8*48*48

__device__ __forceinline__ v8f wmma_f32_16x16x4(v2f a, v2f b, v8f c) {
  // (neg_a, A, neg_b, B, c_mod, C, reuse_a, reuse_b)
  return __builtin_amdgcn_wmma_f32_16x16x4_f32(false, a, false, b, (short)0, c,
                                               false, false);
}

// ---------------------------------------------------------------------------
// Kernel A: qkv = x @ qkv_w.T + qkv_b ; q-part scaled by hd^-0.5
// One wave per 16x16 output tile, K=64 via 16 chained V_WMMA_F32_16X16X4_F32.
// ---------------------------------------------------------------------------
__global__ __launch_bounds__(128) void qkv_gemm_kernel(
    const float* __restrict__ x,     // (NPIX, 64)
    const float* __restrict__ w,     // (192, 64)
    const float* __restrict__ bias,  // (192)
    float* __restrict__ qkv)         // (NPIX, 192)
{
  const int lane = threadIdx.x;                       // 0..31
  const int m0   = blockIdx.x * 16;                   // row tile
  const int n0   = (blockIdx.y * 4 + threadIdx.y) * 16; // col tile (0..176)
  const int l15  = lane & 15;
  const int koff = (lane >> 4) * 2;                   // K sub-offset per half-wave

  const float* aptr = x + (size_t)(m0 + l15) * CH + koff;
  const float* bptr = w + (size_t)(n0 + l15) * CH + koff; // B col N = w row N (W^T)

  v8f acc = {};
#pragma unroll
  for (int k = 0; k < CH; k += 4) {
    v2f a = *(const v2f*)(aptr + k);
    v2f b = *(const v2f*)(bptr + k);
    acc = wmma_f32_16x16x4(a, b, acc);
  }

  const int   ncol  = n0 + l15;
  const float bn    = bias[ncol];
  const float scale = (n0 < CH) ? 0.35355339059327373f : 1.0f; // q * hd^-0.5
#pragma unroll
  for (int r = 0; r < 8; ++r) {
    int row = (lane < 16) ? (m0 + r) : (m0 + 8 + r);
    qkv[(size_t)row * C3 + ncol] = (acc[r] + bn) * scale;
  }
}

// ---------------------------------------------------------------------------
// Kernel B: neighborhood attention, one thread per (pixel, head),
// online softmax over the 7x7 window (v_exp_f32 transcendental path).
// ---------------------------------------------------------------------------
__global__ __launch_bounds__(256) void na2d_attn_kernel(
    const float* __restrict__ qkv,   // (NPIX, 192) = q|k|v
    const float* __restrict__ rpb,   // (8, 13, 13)
    float* __restrict__ out)         // (NPIX, 64)
{
  int t    = blockIdx.x * blockDim.x + threadIdx.x;
  int pix  = t >> 3;
  int head = t & 7;
  if (pix >= NPIX) return;

  int img = pix / IMGPIX;
  int rem = pix - img * IMGPIX;
  int i   = rem / HW;
  int j   = rem - i * HW;
  int ri  = min(max(i - KS / 2, 0), HW - KS);
  int ci  = min(max(j - KS / 2, 0), HW - KS);

  const float* qp = qkv + (size_t)pix * C3 + head * HD;
  float4 q0 = *(const float4*)(qp);
  float4 q1 = *(const float4*)(qp + 4);

  const float* bp   = rpb + head * 169; // 13*13
  const float* base = qkv + (size_t)img * IMGPIX * C3;

  float m = -INFINITY, l = 0.f;
  float acc0 = 0.f, acc1 = 0.f, acc2 = 0.f, acc3 = 0.f;
  float acc4 = 0.f, acc5 = 0.f, acc6 = 0.f, acc7 = 0.f;

  for (int p = 0; p < KS; ++p) {
    int rr  = ri + p;
    int rri = rr - i + (KS - 1);
    const float* rowb = base + (size_t)rr * HW * C3;
    const float* brow = bp + rri * 13 - j + (KS - 1);
#pragma unroll
    for (int qq = 0; qq < KS; ++qq) {
      int cc = ci + qq;
      const float* kp = rowb + (size_t)cc * C3 + CH + head * HD;
      float4 k0 = *(const float4*)(kp);
      float4 k1 = *(const float4*)(kp + 4);
      float s = brow[cc];
      s += q0.x * k0.x + q0.y * k0.y + q0.z * k0.z + q0.w * k0.w
         + q1.x * k1.x + q1.y * k1.y + q1.z * k1.z + q1.w * k1.w;

      float4 v0 = *(const float4*)(kp + CH);      // v = k + 64 floats
      float4 v1 = *(const float4*)(kp + CH + 4);

      float mn = fmaxf(m, s);
      float so = __expf(m - mn);   // 0 on first iter (m = -inf)
      float pw = __expf(s - mn);
      l = l * so + pw;
      acc0 = acc0 * so + pw * v0.x;  acc1 = acc1 * so + pw * v0.y;
      acc2 = acc2 * so + pw * v0.z;  acc3 = acc3 * so + pw * v0.w;
      acc4 = acc4 * so + pw * v1.x;  acc5 = acc5 * so + pw * v1.y;
      acc6 = acc6 * so + pw * v1.z;  acc7 = acc7 * so + pw * v1.w;
      m = mn;
    }
  }

  float inv = 1.0f / l;
  float* op = out + (size_t)pix * CH + head * HD;
  float4 o0 = make_float4(acc0 * inv, acc1 * inv, acc2 * inv, acc3 * inv);
  float4 o1 = make_float4(acc4 * inv, acc5 * inv, acc6 * inv, acc7 * inv);
  *(float4*)(op)     = o0;
  *(float4*)(op + 4) = o1;
}

// ---------------------------------------------------------------------------
// Kernel C: out = LayerNorm(attn @ proj_w.T + proj_b)
// 4 waves cover 16 rows x 64 cols via WMMA, LN fused through LDS.
// ---------------------------------------------------------------------------
__global__ __launch_bounds__(128) void proj_ln_kernel(
    const float* __restrict__ ain,   // (NPIX, 64)
    const float* __restrict__ w,     // (64, 64)
    const float* __restrict__ bias,  // (64)
    const float* __restrict__ g,     // (64)
    const float* __restrict__ be,    // (64)
    float* __restrict__ out)         // (NPIX, 64)
{
  __shared__ float tile[16 * 68];    // stride 68 -> conflict-free column reads
  __shared__ float mu_s[16];
  __shared__ float rs_s[16];

  const int lane = threadIdx.x;
  const int wv   = threadIdx.y;      // 0..3 -> col tile
  const int m0   = blockIdx.x * 16;
  const int n0   = wv * 16;
  const int l15  = lane & 15;
  const int koff = (lane >> 4) * 2;

  const float* aptr = ain + (size_t)(m0 + l15) * CH + koff;
  const float* bptr = w + (size_t)(n0 + l15) * CH + koff;

  v8f acc = {};
#pragma unroll
  for (int k = 0; k < CH; k += 4) {
    v2f a = *(const v2f*)(aptr + k);
    v2f b = *(const v2f*)(bptr + k);
    acc = wmma_f32_16x16x4(a, b, acc);
  }

  const int   ncol = n0 + l15;
  const float bn   = bias[ncol];
#pragma unroll
  for (int r = 0; r < 8; ++r) {
    int row = (lane < 16) ? r : (8 + r);
    tile[row * 68 + ncol] = acc[r] + bn;
  }
  __syncthreads();

  int tid = wv * 32 + lane;          // 0..127
  if (tid < 16) {
    float s = 0.f, s2 = 0.f;
#pragma unroll
    for (int e = 0; e < CH; ++e) {
      float v = tile[tid * 68 + e];
      s += v;
      s2 += v * v;
    }
    float mu  = s * (1.0f / CH);
    float var = s2 * (1.0f / CH) - mu * mu;
    mu_s[tid] = mu;
    rs_s[tid] = rsqrtf(var + 1e-5f);
  }
  __syncthreads();

  int row = tid >> 3;                // 0..15
  int c0  = (tid & 7) * 8;
  float mu = mu_s[row], rs = rs_s[row];
  float* op = out + (size_t)(m0 + row) * CH + c0;
#pragma unroll
  for (int e = 0; e < 8; ++e) {
    int c = c0 + e;
    op[e] = (tile[row * 68 + c] - mu) * rs * g[c] + be[c];
  }
}

// ---------------------------------------------------------------------------
extern "C" void kernel_launch(void* const* d_in, const int* in_sizes, int n_in,
                              void* d_out, int out_size, void* d_ws, size_t ws_size,
                              hipStream_t stream) {
  const float* x      = (const float*)d_in[0];
  const float* qkv_w  = (const float*)d_in[1];
  const float* qkv_b  = (const float*)d_in[2];
  const float* proj_w = (const float*)d_in[3];
  const float* proj_b = (const float*)d_in[4];
  const float* rpb    = (const float*)d_in[5];
  const float* ln_g   = (const float*)d_in[6];
  const float* ln_b   = (const float*)d_in[7];
  float* out = (float*)d_out;

  float* qkv  = (float*)d_ws;                  // NPIX*192 floats (28.3 MB)
  float* attn = qkv + (size_t)NPIX * C3;       // NPIX*64  floats ( 9.4 MB)

  dim3 blkA(32, 4);
  dim3 grdA(NPIX / 16, C3 / 64);               // 2304 x 3 (4 col tiles/block)
  qkv_gemm_kernel<<<grdA, blkA, 0, stream>>>(x, qkv_w, qkv_b, qkv);

  na2d_attn_kernel<<<(NPIX * HEADS) / 256, 256, 0, stream>>>(qkv, rpb, attn);

  dim3 blkC(32, 4);
  proj_ln_kernel<<<NPIX / 16, blkC, 0, stream>>>(attn, proj_w, proj_b,
                                                 ln_g, ln_b, out);
}